// BoundaryLoss_4286377362141
// MI455X (gfx1250) — compile-verified
//
#include <hip/hip_runtime.h>
#include <hip/hip_bf16.h>
#include <math.h>

#define BATCH 8
#define HH 256
#define WW 256
#define HWSZ (HH * WW)
#define NTOT (BATCH * HWSZ)
#define BIGD 512            // H + W, reference clamp

typedef __attribute__((ext_vector_type(2))) float v2f;
typedef __attribute__((ext_vector_type(8))) float v8f;

static __device__ __forceinline__ int imin(int a, int b) { return a < b ? a : b; }

// ---- CDNA5 async global->LDS staging (ASYNCcnt path) -----------------------
static __device__ __forceinline__ void async_g2lds_b32(const float* gsrc, float* lds_dst) {
#if __has_builtin(__builtin_amdgcn_global_load_async_to_lds_b32)
  __builtin_amdgcn_global_load_async_to_lds_b32(
      (__attribute__((address_space(1))) int*)(int*)gsrc,
      (__attribute__((address_space(3))) int*)(int*)lds_dst,
      /*imm offset*/ 0, /*cpol*/ 0);
#else
  unsigned loff = (unsigned)(unsigned long long)(__attribute__((address_space(3))) char*)lds_dst;
  asm volatile("global_load_async_to_lds_b32 %0, %1, off"
               :: "v"(loff), "v"(gsrc) : "memory");
#endif
}

static __device__ __forceinline__ void wait_async0() {
#if __has_builtin(__builtin_amdgcn_s_wait_asynccnt)
  __builtin_amdgcn_s_wait_asynccnt(0);
#else
  asm volatile("s_wait_asynccnt 0" ::: "memory");
#endif
}

// Single-DS-op XOR fold within wave32 (all lanes active): ds_swizzle_b32
// group-of-32 mode, offset = {xor_mask<<10 | or_mask<<5 | and_mask}.
#define XOR_FOLD_ADD(s, m)                                                    \
  (s) += __int_as_float(__builtin_amdgcn_ds_swizzle(                          \
      __float_as_int(s), (((m) << 10) | 0x1f)))

// ---- Pass 1: vertical two-way 1-D EDT, writes clamped g^2 for both classes --
// One thread per (b,w) column. Mask bits packed from streaming (pipelined)
// loads; down-scan spilled to LDS (same-thread reuse, no barrier needed).
__global__ void __launch_bounds__(64)
edt_vert_kernel(const float* __restrict__ targets,
                float* __restrict__ g2pos,   // _edt(~fg)^2 after vertical pass
                float* __restrict__ g2neg,   // _edt( fg)^2 after vertical pass
                float* __restrict__ out) {
  __shared__ unsigned sdown[HH * 64];        // 64 KB: packed {dneg:u16, dpos:u16}
  const int tid = threadIdx.x;
  const int c   = blockIdx.x * 64 + tid;     // 0 .. B*W-1
  const int b   = c >> 8;
  const int w   = c & (WW - 1);
  if (blockIdx.x == 0 && tid == 0) out[0] = 0.0f;   // zero the accumulator

  const float* tp = targets + (size_t)b * HWSZ + w;

  // Pack fg bits for the whole column; loads are address-independent so the
  // compiler can keep ~32 in flight (kills the scan's load-latency chain).
  unsigned bits[8];
#pragma unroll
  for (int c32 = 0; c32 < 8; ++c32) {
    float v[32];
#pragma unroll
    for (int i = 0; i < 32; ++i) v[i] = tp[(size_t)(c32 * 32 + i) * WW];
    unsigned m = 0;
#pragma unroll
    for (int i = 0; i < 32; ++i) m |= (v[i] > 0.5f) ? (1u << i) : 0u;
    bits[c32] = m;
  }

  // Down scan: exact integer distance counters, clamp at BIGD (matches
  // reference init=H+W then min(g,big)).
  int dp = BIGD, dn = BIGD;
#pragma unroll 4
  for (int h = 0; h < HH; ++h) {
    const bool fg = (bits[h >> 5] >> (h & 31)) & 1u;
    dp = fg ? dp + 1 : 0;          // _edt(~fg): zero where ~fg
    dn = fg ? 0 : dn + 1;          // _edt( fg): zero where fg
    const unsigned pdp = (unsigned)imin(dp, BIGD);
    const unsigned pdn = (unsigned)imin(dn, BIGD);
    sdown[h * 64 + tid] = pdp | (pdn << 16);
  }

  // Up scan fused with min + squared store (coalesced across lanes).
  int up = BIGD, un = BIGD;
#pragma unroll 4
  for (int h = HH - 1; h >= 0; --h) {
    const bool fg = (bits[h >> 5] >> (h & 31)) & 1u;
    up = fg ? up + 1 : 0;
    un = fg ? 0 : un + 1;
    const unsigned pk = sdown[h * 64 + tid];
    const int gp = imin(imin(up, BIGD), (int)(pk & 0xffffu));
    const int gn = imin(imin(un, BIGD), (int)(pk >> 16));
    const float fgp = (float)gp, fgn = (float)gn;
    const size_t idx = (size_t)b * HWSZ + (size_t)h * WW + w;
    g2pos[idx] = fgp * fgp;
    g2neg[idx] = fgn * fgn;
  }
}

// ---- Pass 2: dense min-plus over columns + weighted BCE + reduction --------
// One block per row (b,h); g^2 rows async-staged to LDS; broadcast float4
// reads feed 4 fma+min per iteration. Reduction: fp32 WMMA (ones-A column
// sums) + 4 single-op ds_swizzle folds, then cross-wave LDS + atomicAdd.
__global__ void __launch_bounds__(256)
edt_horiz_loss_kernel(const float* __restrict__ pred,
                      const float* __restrict__ targets,
                      const float* __restrict__ g2pos,
                      const float* __restrict__ g2neg,
                      float* __restrict__ out) {
  __shared__ float sp[WW];
  __shared__ float sn[WW];
  __shared__ float wsum[8];
  const int tid  = threadIdx.x;          // == w
  const int row  = blockIdx.x;           // b*H + h
  const size_t base = (size_t)row * WW;

  async_g2lds_b32(g2pos + base + tid, sp + tid);
  async_g2lds_b32(g2neg + base + tid, sn + tid);

  const float x = pred[base + tid];
  const float t = targets[base + tid];

  wait_async0();
  __syncthreads();

  const bool fg = t > 0.5f;
  const float4* g4 = (const float4*)(fg ? sp : sn);

  float d2 = 3.4e38f;
  float fd = (float)tid;                 // fd = w - w', decremented per w'
#pragma unroll 4
  for (int j = 0; j < WW / 4; ++j) {
    const float4 v = g4[j];              // broadcast ds_load_b128
    d2 = fminf(d2, fmaf(fd, fd, v.x)); fd -= 1.0f;
    d2 = fminf(d2, fmaf(fd, fd, v.y)); fd -= 1.0f;
    d2 = fminf(d2, fmaf(fd, fd, v.z)); fd -= 1.0f;
    d2 = fminf(d2, fmaf(fd, fd, v.w)); fd -= 1.0f;
  }

  const float dist = sqrtf(d2);
  const float wgt  = 1.0f / (1.0f + expf((dist - 3.0f) * 0.2f)); // sigmoid(-(d-3)/5)
  const float bce  = fmaxf(x, 0.0f) - x * t + log1pf(expf(-fabsf(x)));
  const float val  = bce * wgt * (1.0f / (float)NTOT);

  // ---- wave32 sum; fp32 WMMA does the first fold exactly ----
#if __has_builtin(__builtin_amdgcn_wmma_f32_16x16x4_f32)
  v2f A;  A[0] = 1.0f; A[1] = 1.0f;      // 16x4 all-ones
  v2f Bm; Bm[0] = val; Bm[1] = 0.0f;     // 4x16 carrying per-lane partials
  v8f C = {};
  v8f D = __builtin_amdgcn_wmma_f32_16x16x4_f32(
      false, A, false, Bm, (short)0, C, false, false);
  float s = D[0];                        // lane L holds colsum(L%16)=val(n)+val(n+16)
  XOR_FOLD_ADD(s, 1);
  XOR_FOLD_ADD(s, 2);
  XOR_FOLD_ADD(s, 4);
  XOR_FOLD_ADD(s, 8);
#else
  float s = val;
  s += __shfl_xor(s, 16);
  s += __shfl_xor(s, 8);
  s += __shfl_xor(s, 4);
  s += __shfl_xor(s, 2);
  s += __shfl_xor(s, 1);
#endif
  if ((tid & 31) == 0) wsum[tid >> 5] = s;
  __syncthreads();
  if (tid == 0) {
    float bsum = 0.0f;
#pragma unroll
    for (int i = 0; i < 8; ++i) bsum += wsum[i];
    atomicAdd(out, bsum);
  }
}

extern "C" void kernel_launch(void* const* d_in, const int* in_sizes, int n_in,
                              void* d_out, int out_size, void* d_ws, size_t ws_size,
                              hipStream_t stream) {
  (void)in_sizes; (void)n_in; (void)out_size; (void)ws_size;
  const float* pred = (const float*)d_in[0];
  const float* targ = (const float*)d_in[1];
  float* out   = (float*)d_out;
  float* g2pos = (float*)d_ws;           // NTOT floats
  float* g2neg = g2pos + NTOT;           // NTOT floats (4 MB total workspace)

  edt_vert_kernel<<<(BATCH * WW) / 64, 64, 0, stream>>>(targ, g2pos, g2neg, out);
  edt_horiz_loss_kernel<<<BATCH * HH, 256, 0, stream>>>(pred, targ, g2pos, g2neg, out);
}